// GCN_66640712565428
// MI455X (gfx1250) — compile-verified
//
#include <hip/hip_runtime.h>
#include <hip/hip_bf16.h>

// ---------------------------------------------------------------------------
// GCN (3x GCNConv) for MI455X / gfx1250, wave32.
// GEMMs use V_WMMA_F32_16X16X4_F32 (exact f32, matches reference precision).
// Scatter-add uses hardware global_atomic_add_f32.
// ---------------------------------------------------------------------------

#define NNODES 50000
#define NEDGES 800000
#define NMSG   (NEDGES + NNODES)

typedef __attribute__((ext_vector_type(2))) float v2f;
typedef __attribute__((ext_vector_type(8))) float v8f;

// ---------------------------------------------------------------------------
// Degree accumulation: deg[dst[e]] += 1 for the E real edges.
// ---------------------------------------------------------------------------
__global__ __launch_bounds__(256) void deg_kernel(const long long* __restrict__ dst,
                                                  float* __restrict__ deg, int nE) {
    int e = blockIdx.x * blockDim.x + threadIdx.x;
    if (e < nE) {
        unsafeAtomicAdd(&deg[(int)dst[e]], 1.0f);
    }
}

// dinv[i] = rsqrt(deg[i] + 1)   (+1 for the self loop; always >= 1)
__global__ __launch_bounds__(256) void dinv_kernel(float* __restrict__ deg, int n) {
    int i = blockIdx.x * blockDim.x + threadIdx.x;
    if (i < n) {
        deg[i] = rsqrtf(deg[i] + 1.0f);
    }
}

// norm[e] for all E+N messages (self-loops appended at the end).
__global__ __launch_bounds__(256) void norm_kernel(const long long* __restrict__ src,
                                                   const long long* __restrict__ dst,
                                                   const float* __restrict__ dinv,
                                                   float* __restrict__ norm,
                                                   int nE, int nN) {
    int e = blockIdx.x * blockDim.x + threadIdx.x;
    if (e >= nE + nN) return;
    if (e < nE) {
        norm[e] = dinv[(int)src[e]] * dinv[(int)dst[e]];
    } else {
        float v = dinv[e - nE];
        norm[e] = v * v;
    }
}

// ---------------------------------------------------------------------------
// WMMA f32 GEMM: H[M x NW] = X[M x K] @ W[K x NW], row-major.
// One wave (32 lanes) computes one 16x16 tile; K-loop in steps of 4 using
// V_WMMA_F32_16X16X4_F32. M = 50000 is an exact multiple of 16 (no M guard).
// NGUARD handles NW=40 (last N-tile covers cols 32..47, valid < 40).
//
// Fragment layouts (ISA 7.12.2, 32-bit, wave32):
//   A 16x4 : lane l -> row m = l&15 ; VGPR0 = K=2*(l>>4), VGPR1 = K=2*(l>>4)+1
//   B 4x16 : lane l -> col n = l&15 ; VGPR0 = K=2*(l>>4), VGPR1 = K=2*(l>>4)+1
//   C 16x16: VGPR r, lane l -> (m = r + 8*(l>>4), n = l&15)
// ---------------------------------------------------------------------------
template <int K, int NW, bool NGUARD>
__global__ __launch_bounds__(32) void gemm_wmma_f32(const float* __restrict__ X,
                                                    const float* __restrict__ W,
                                                    float* __restrict__ H) {
    const int m0   = blockIdx.x * 16;
    const int n0   = blockIdx.y * 16;
    const int lane = threadIdx.x;        // 0..31
    const int mr   = lane & 15;
    const int kb   = (lane >> 4) * 2;    // 0 or 2
    const int nn   = n0 + (lane & 15);
    const bool nok = !NGUARD || (nn < NW);

    const float* __restrict__ xrow = X + (size_t)(m0 + mr) * K;

    v8f c = {0.f, 0.f, 0.f, 0.f, 0.f, 0.f, 0.f, 0.f};

    #pragma unroll 4
    for (int k0 = 0; k0 < K; k0 += 4) {
        v2f a, b;
        a[0] = xrow[k0 + kb];
        a[1] = xrow[k0 + kb + 1];
        b[0] = nok ? W[(size_t)(k0 + kb) * NW + nn] : 0.0f;
        b[1] = nok ? W[(size_t)(k0 + kb + 1) * NW + nn] : 0.0f;
        c = __builtin_amdgcn_wmma_f32_16x16x4_f32(
                /*neg_a=*/false, a, /*neg_b=*/false, b,
                /*c_mod=*/(short)0, c, /*reuse_a=*/false, /*reuse_b=*/false);
    }

    if (nok) {
        const int mbase = m0 + ((lane >> 4) * 8);
        #pragma unroll
        for (int r = 0; r < 8; ++r) {
            H[(size_t)(mbase + r) * NW + nn] = c[r];
        }
    }
}

// ---------------------------------------------------------------------------
// Scatter: out[dst[m]] += H[src[m]] * norm[m], 4 floats per thread.
// Messages m in [0,E) are real edges; m in [E, E+N) are self loops.
// 16 (F=64) or 10 (F=40) consecutive threads share one message -> coalesced
// gather of the source row; atomics go to hardware global_atomic_add_f32.
// ---------------------------------------------------------------------------
template <int F>
__global__ __launch_bounds__(256) void scatter_kernel(const float* __restrict__ H,
                                                      const float* __restrict__ norm,
                                                      const long long* __restrict__ src,
                                                      const long long* __restrict__ dst,
                                                      float* __restrict__ out,
                                                      long long total) {
    constexpr int F4 = F / 4;
    long long t = (long long)blockIdx.x * blockDim.x + threadIdx.x;
    if (t >= total) return;
    int msg = (int)(t / F4);
    int c   = (int)(t % F4) * 4;
    int s, d;
    if (msg < NEDGES) {
        s = (int)src[msg];
        d = (int)dst[msg];
    } else {
        s = d = msg - NEDGES;
    }
    float nv = norm[msg];
    const float4 v = *(const float4*)(H + (size_t)s * F + c);
    float* op = out + (size_t)d * F + c;
    unsafeAtomicAdd(op + 0, v.x * nv);
    unsafeAtomicAdd(op + 1, v.y * nv);
    unsafeAtomicAdd(op + 2, v.z * nv);
    unsafeAtomicAdd(op + 3, v.w * nv);
}

// ---------------------------------------------------------------------------
// Bias (+ optional ReLU), in place over [NNODES x F].
// ---------------------------------------------------------------------------
template <int F, bool RELU>
__global__ __launch_bounds__(256) void bias_act_kernel(float* __restrict__ A,
                                                       const float* __restrict__ bias,
                                                       long long total) {
    long long t = (long long)blockIdx.x * blockDim.x + threadIdx.x;
    if (t >= total) return;
    int c = (int)(t % F);
    float v = A[t] + bias[c];
    if (RELU) v = fmaxf(v, 0.0f);
    A[t] = v;
}

// ---------------------------------------------------------------------------
static inline size_t align256(size_t x) { return (x + 255) & ~(size_t)255; }

extern "C" void kernel_launch(void* const* d_in, const int* in_sizes, int n_in,
                              void* d_out, int out_size, void* d_ws, size_t ws_size,
                              hipStream_t stream) {
    const float*     x   = (const float*)d_in[0];
    const long long* ei  = (const long long*)d_in[1];   // int64 [2, E]
    const float*     W1  = (const float*)d_in[2];
    const float*     b1  = (const float*)d_in[3];
    const float*     W2  = (const float*)d_in[4];
    const float*     b2  = (const float*)d_in[5];
    const float*     W3  = (const float*)d_in[6];
    const float*     b3  = (const float*)d_in[7];
    float*           out = (float*)d_out;               // [N, 40]

    const long long* src = ei;            // edge_index[0]
    const long long* dst = ei + NEDGES;   // edge_index[1]

    // ---- workspace carve (~29.2 MB) ----
    char*  ws   = (char*)d_ws;
    size_t off  = 0;
    float* dinv = (float*)(ws + off); off += align256((size_t)NNODES * 4);
    float* norm = (float*)(ws + off); off += align256((size_t)NMSG * 4);
    float* bufh = (float*)(ws + off); off += align256((size_t)NNODES * 64 * 4);
    float* bufa = (float*)(ws + off); off += align256((size_t)NNODES * 64 * 4);
    (void)ws_size; (void)n_in; (void)in_sizes; (void)out_size;

    // ---- normalization coefficients ----
    hipMemsetAsync(dinv, 0, (size_t)NNODES * 4, stream);
    deg_kernel<<<(NEDGES + 255) / 256, 256, 0, stream>>>(dst, dinv, NEDGES);
    dinv_kernel<<<(NNODES + 255) / 256, 256, 0, stream>>>(dinv, NNODES);
    norm_kernel<<<(NMSG + 255) / 256, 256, 0, stream>>>(src, dst, dinv, norm,
                                                        NEDGES, NNODES);

    const int MT = NNODES / 16;  // 3125, exact

    // ---- layer 1: h = relu(agg(x @ W1) + b1) ----
    gemm_wmma_f32<128, 64, false><<<dim3(MT, 4), 32, 0, stream>>>(x, W1, bufh);
    hipMemsetAsync(bufa, 0, (size_t)NNODES * 64 * 4, stream);
    {
        long long total = (long long)NMSG * (64 / 4);
        scatter_kernel<64><<<(int)((total + 255) / 256), 256, 0, stream>>>(
            bufh, norm, src, dst, bufa, total);
        long long tb = (long long)NNODES * 64;
        bias_act_kernel<64, true><<<(int)((tb + 255) / 256), 256, 0, stream>>>(
            bufa, b1, tb);
    }

    // ---- layer 2: h = relu(agg(h @ W2) + b2) ----
    gemm_wmma_f32<64, 64, false><<<dim3(MT, 4), 32, 0, stream>>>(bufa, W2, bufh);
    hipMemsetAsync(bufa, 0, (size_t)NNODES * 64 * 4, stream);   // GEMM2 consumed bufa
    {
        long long total = (long long)NMSG * (64 / 4);
        scatter_kernel<64><<<(int)((total + 255) / 256), 256, 0, stream>>>(
            bufh, norm, src, dst, bufa, total);
        long long tb = (long long)NNODES * 64;
        bias_act_kernel<64, true><<<(int)((tb + 255) / 256), 256, 0, stream>>>(
            bufa, b2, tb);
    }

    // ---- layer 3: out = agg(h @ W3) + b3 ----
    gemm_wmma_f32<64, 40, true><<<dim3(MT, 3), 32, 0, stream>>>(bufa, W3, bufh);
    hipMemsetAsync(out, 0, (size_t)NNODES * 40 * 4, stream);
    {
        long long total = (long long)NMSG * (40 / 4);
        scatter_kernel<40><<<(int)((total + 255) / 256), 256, 0, stream>>>(
            bufh, norm, src, dst, out, total);
        long long tb = (long long)NNODES * 40;
        bias_act_kernel<40, false><<<(int)((tb + 255) / 256), 256, 0, stream>>>(
            out, b3, tb);
    }
}